// TemporalGraphUNet_87797721464866
// MI455X (gfx1250) — compile-verified
//
#include <hip/hip_runtime.h>
#include <math.h>

// ---------------------------------------------------------------------------
// TemporalGraphUNet on MI455X (gfx1250), all GEMMs via V_WMMA_F32_16X16X4_F32.
// N=4096 nodes, H=256, C_IN=C_OUT=64, DEPTH=3, RATIO=0.5.
//
// Dominant work: dense augment matmul A0@A0 (2*4096^3 ~ 137 GFLOP fp32);
// operands are 64MB each and resident in the 192MB L2 -> compute bound on the
// f32 matrix pipe. fp32 WMMA is chosen deliberately: adjacency entries are
// exact small integers and fp32 accumulation of integer products is EXACT
// below 2^24 (order-independent, matches the reference); bf16 inputs would
// break exactness once level-2 path counts exceed 256.
//
// GEMM: 128 threads / 4 waves, block tile 128x64, wave tile 64x32
// (4x2 v8f accumulators = 32 WMMA per 16-wide K-tile vs 24 LDS fragment
// loads), register double-buffered global->LDS staging.
// All GEMM dims in this pipeline are multiples of {M:128, N:64, K:16}.
// ---------------------------------------------------------------------------

#define N_NODES 4096
#define H_DIM   256
#define C_IN    64
#define C_OUT   64
#define DEPTH_C 3

typedef float v2f __attribute__((ext_vector_type(2)));
typedef float v8f __attribute__((ext_vector_type(8)));

// ---------------------------------------------------------------------------
// Generic f32 WMMA GEMM:  C[M,N] = opA(A) @ opB(B)
//   transA: element (m,k) = A[k*lda + m]   (else A[m*lda + k])
//   transB: element (k,n) = B[n*ldb + k]   (else B[k*ldb + n])
//   diagA/diagB: operand element at global (r,c) with r==c forced to 1.0
//                (implements _augment's  A*(1-I)+I  without a copy)
//   zeroDiagC: output diagonal forced to 0 (augment's final mask)
// ---------------------------------------------------------------------------
__global__ __launch_bounds__(128)
void tgun_wmma_gemm_f32(const float* __restrict__ A, const float* __restrict__ B,
                        float* __restrict__ C,
                        int M, int N, int K, int lda, int ldb, int ldc,
                        int transA, int transB, int diagA, int diagB, int zeroDiagC)
{
    __shared__ float As[128][17];  // 128 x 16 (+1 pad)
    __shared__ float Bs[16][65];   // 16 x 64  (+1 pad)

    const int tid    = threadIdx.x;
    const int wave   = tid >> 5;
    const int lane   = tid & 31;
    const int laneM  = lane & 15;
    const int laneHi = lane >> 4;          // 0: lanes 0-15, 1: lanes 16-31
    const int row0   = blockIdx.y * 128;
    const int col0   = blockIdx.x * 64;
    const int wr     = (wave >> 1) * 64;   // wave row offset within block tile
    const int wc     = (wave & 1) * 32;    // wave col offset

    v8f acc[4][2] = {};

    const int ntiles = K >> 4;

    // ---- stage tile 0 directly into LDS ----
    {
        for (int j = 0; j < 16; ++j) {                 // A: 128x16 = 2048 elems
            int idx = tid + j * 128;
            int r = idx >> 4, kk = idx & 15;
            int gr = row0 + r, gk = kk;
            float v = transA ? A[(size_t)gk * lda + gr] : A[(size_t)gr * lda + gk];
            if (diagA && gr == gk) v = 1.0f;
            As[r][kk] = v;
        }
        for (int j = 0; j < 8; ++j) {                  // B: 16x64 = 1024 elems
            int idx = tid + j * 128;
            int kk = idx >> 6, c = idx & 63;
            int gk = kk, gc = col0 + c;
            float v = transB ? B[(size_t)gc * ldb + gk] : B[(size_t)gk * ldb + gc];
            if (diagB && gk == gc) v = 1.0f;
            Bs[kk][c] = v;
        }
    }
    __syncthreads();

    float aStage[16], bStage[8];

    for (int t = 0; t < ntiles; ++t) {
        // ---- prefetch tile t+1 from global into registers ----
        const bool havenext = (t + 1 < ntiles);
        if (havenext) {
            const int k0n = (t + 1) << 4;
            for (int j = 0; j < 16; ++j) {
                int idx = tid + j * 128;
                int r = idx >> 4, kk = idx & 15;
                int gr = row0 + r, gk = k0n + kk;
                float v = transA ? A[(size_t)gk * lda + gr] : A[(size_t)gr * lda + gk];
                if (diagA && gr == gk) v = 1.0f;
                aStage[j] = v;
            }
            for (int j = 0; j < 8; ++j) {
                int idx = tid + j * 128;
                int kk = idx >> 6, c = idx & 63;
                int gk = k0n + kk, gc = col0 + c;
                float v = transB ? B[(size_t)gc * ldb + gk] : B[(size_t)gk * ldb + gc];
                if (diagB && gk == gc) v = 1.0f;
                bStage[j] = v;
            }
        }

        // ---- compute current K-tile: 4 k-steps of 4, 8 WMMA each ----
        for (int ks = 0; ks < 16; ks += 4) {
            const int kb = ks + laneHi * 2;  // lanes 0-15: K0/K1, lanes 16-31: K2/K3
            v2f afr[4], bfr[2];
            for (int mi = 0; mi < 4; ++mi) {
                int r = wr + mi * 16 + laneM;
                afr[mi].x = As[r][kb];
                afr[mi].y = As[r][kb + 1];
            }
            for (int ni = 0; ni < 2; ++ni) {
                int c = wc + ni * 16 + laneM;
                bfr[ni].x = Bs[kb][c];
                bfr[ni].y = Bs[kb + 1][c];
            }
            for (int mi = 0; mi < 4; ++mi)
                for (int ni = 0; ni < 2; ++ni)
                    acc[mi][ni] = __builtin_amdgcn_wmma_f32_16x16x4_f32(
                        false, afr[mi], false, bfr[ni], (short)0, acc[mi][ni],
                        false, false);
        }
        __syncthreads();

        // ---- commit staged tile t+1 into LDS ----
        if (havenext) {
            for (int j = 0; j < 16; ++j) {
                int idx = tid + j * 128;
                As[idx >> 4][idx & 15] = aStage[j];
            }
            for (int j = 0; j < 8; ++j) {
                int idx = tid + j * 128;
                Bs[idx >> 6][idx & 63] = bStage[j];
            }
        }
        __syncthreads();
    }

    // Store: C/D layout -> VGPR r: lanes 0-15 = (M=r, N=lane), lanes 16-31 = (M=r+8)
    for (int mi = 0; mi < 4; ++mi)
        for (int ni = 0; ni < 2; ++ni) {
            int gc = col0 + wc + ni * 16 + laneM;
            for (int r = 0; r < 8; ++r) {
                int gr = row0 + wr + mi * 16 + r + laneHi * 8;
                float v = acc[mi][ni][r];
                if (zeroDiagC && gr == gc) v = 0.0f;
                C[(size_t)gr * ldc + gc] = v;
            }
        }
}

// ---------------------------------------------------------------------------
// Small elementwise / graph kernels
// ---------------------------------------------------------------------------
__global__ void tgun_scatter_edges(const int* __restrict__ ei, float* __restrict__ A,
                                   int E, int n)
{
    int e = blockIdx.x * blockDim.x + threadIdx.x;
    if (e >= E) return;
    int src = ei[e];
    int dst = ei[E + e];
    atomicAdd(&A[(size_t)src * n + dst], 1.0f);  // +1.0 each -> exact, order-free
}

// dis[j] = rsqrt(2 + colsum(A)[j])   (improved GCN: A_hat = A + 2I)
__global__ void tgun_coldeg_dis(const float* __restrict__ A, float* __restrict__ dis, int n)
{
    int j = blockIdx.x * blockDim.x + threadIdx.x;
    if (j >= n) return;
    float s = 2.0f;
    for (int i = 0; i < n; ++i) s += A[(size_t)i * n + j];
    dis[j] = (s > 0.0f) ? rsqrtf(s) : 0.0f;
}

__global__ void tgun_rowscale(const float* __restrict__ X, const float* __restrict__ dis,
                              float* __restrict__ Y, int n, int c)
{
    size_t idx = (size_t)blockIdx.x * blockDim.x + threadIdx.x;
    if (idx >= (size_t)n * c) return;
    Y[idx] = X[idx] * dis[idx / c];
}

// out = act( dis_row * (Z + 2*Ys) + b_col )
__global__ void tgun_gcn_epi(const float* __restrict__ Z, const float* __restrict__ Ys,
                             const float* __restrict__ dis, const float* __restrict__ b,
                             float* __restrict__ out, int n, int c, int do_relu)
{
    size_t idx = (size_t)blockIdx.x * blockDim.x + threadIdx.x;
    if (idx >= (size_t)n * c) return;
    int row = (int)(idx / c);
    int col = (int)(idx % c);
    float v = dis[row] * (Z[idx] + 2.0f * Ys[idx]) + b[col];
    out[idx] = do_relu ? fmaxf(v, 0.0f) : v;
}

__global__ void tgun_pnorm(const float* __restrict__ p, float* __restrict__ out, int h)
{
    __shared__ float red[256];
    int t = threadIdx.x;
    float v = (t < h) ? p[t] * p[t] : 0.0f;
    red[t] = v;
    __syncthreads();
    for (int s = 128; s > 0; s >>= 1) {
        if (t < s) red[t] += red[t + s];
        __syncthreads();
    }
    if (t == 0) out[0] = sqrtf(red[0]);
}

__global__ void tgun_score(const float* __restrict__ x, const float* __restrict__ p,
                           const float* __restrict__ pn, float* __restrict__ sc,
                           int n, int h)
{
    int j = blockIdx.x * blockDim.x + threadIdx.x;
    if (j >= n) return;
    const float* xr = x + (size_t)j * h;
    float d = 0.0f;
    for (int c = 0; c < h; ++c) d += xr[c] * p[c];
    sc[j] = tanhf(d / pn[0]);
}

// Stable descending rank = jax.lax.top_k order; ranks form a permutation.
__global__ void tgun_rank_topk(const float* __restrict__ s, int* __restrict__ perm,
                               int n, int k)
{
    int j = blockIdx.x * blockDim.x + threadIdx.x;
    if (j >= n) return;
    float sj = s[j];
    int r = 0;
    for (int l = 0; l < n; ++l) {
        float sl = s[l];
        r += (sl > sj) || (sl == sj && l < j);
    }
    if (r < k) perm[r] = j;
}

__global__ void tgun_gather_x(const float* __restrict__ x, const float* __restrict__ sc,
                              const int* __restrict__ perm, float* __restrict__ xk,
                              int k, int h)
{
    size_t idx = (size_t)blockIdx.x * blockDim.x + threadIdx.x;
    if (idx >= (size_t)k * h) return;
    int r = (int)(idx / h);
    int c = (int)(idx % h);
    int pr = perm[r];
    xk[idx] = x[(size_t)pr * h + c] * sc[pr];
}

__global__ void tgun_gather_A(const float* __restrict__ Asrc, const int* __restrict__ perm,
                              float* __restrict__ Ak, int k, int n)
{
    size_t idx = (size_t)blockIdx.x * blockDim.x + threadIdx.x;
    if (idx >= (size_t)k * k) return;
    int r = (int)(idx / k);
    int c = (int)(idx % k);
    Ak[idx] = Asrc[(size_t)perm[r] * n + perm[c]];
}

__global__ void tgun_copy(const float* __restrict__ src, float* __restrict__ dst, size_t n)
{
    size_t idx = (size_t)blockIdx.x * blockDim.x + threadIdx.x;
    if (idx < n) dst[idx] = src[idx];
}

// xnew[perm[r], :] += xl[r, :]   (perm entries unique -> no atomics)
__global__ void tgun_unpool_add(const int* __restrict__ perm, const float* __restrict__ xl,
                                float* __restrict__ xnew, int k, int h)
{
    size_t idx = (size_t)blockIdx.x * blockDim.x + threadIdx.x;
    if (idx >= (size_t)k * h) return;
    int r = (int)(idx / h);
    int c = (int)(idx % h);
    xnew[(size_t)perm[r] * h + c] += xl[idx];
}

__device__ __forceinline__ float tgun_sigmoid(float x) { return 1.0f / (1.0f + expf(-x)); }

// Persistent sequential LSTM: Gx = xk @ Wih^T precomputed by WMMA GEMM.
// One 1024-thread WG; h/c/z in LDS; per step only the 1024x256 recurrent matvec.
__global__ __launch_bounds__(1024)
void tgun_lstm_scan(const float* __restrict__ Gx, const float* __restrict__ Whh,
                    const float* __restrict__ bih, const float* __restrict__ bhh,
                    float* __restrict__ ys, int steps)
{
    __shared__ float h[H_DIM];
    __shared__ float cst[H_DIM];
    __shared__ float z[4 * H_DIM];
    int t = threadIdx.x;
    if (t < H_DIM) { h[t] = 0.0f; cst[t] = 0.0f; }
    __syncthreads();
    for (int s = 0; s < steps; ++s) {
        float acc = Gx[(size_t)s * (4 * H_DIM) + t] + bih[t] + bhh[t];
        const float* w = Whh + (size_t)t * H_DIM;
        for (int u = 0; u < H_DIM; ++u) acc += w[u] * h[u];
        z[t] = acc;
        __syncthreads();
        if (t < H_DIM) {
            float zi = z[t], zf = z[H_DIM + t], zg = z[2 * H_DIM + t], zo = z[3 * H_DIM + t];
            float cc = tgun_sigmoid(zf) * cst[t] + tgun_sigmoid(zi) * tanhf(zg);
            float hh = tgun_sigmoid(zo) * tanhf(cc);
            cst[t] = cc; h[t] = hh;
            ys[(size_t)s * H_DIM + t] = hh;
        }
        __syncthreads();
    }
}

// ---------------------------------------------------------------------------
// Host-side orchestration
// ---------------------------------------------------------------------------
static inline dim3 gsz(size_t n) { return dim3((unsigned)((n + 255) / 256)); }

struct TgunCtx {
    hipStream_t stream;
    float *T0, *T1, *DIS;
};

static void tgun_gemm(hipStream_t stream, const float* A, const float* B, float* C,
                      int M, int N, int K, int lda, int ldb, int ldc,
                      int tA, int tB, int dA, int dB, int zC)
{
    dim3 grid(N / 64, M / 128);
    tgun_wmma_gemm_f32<<<grid, dim3(128), 0, stream>>>(A, B, C, M, N, K, lda, ldb, ldc,
                                                       tA, tB, dA, dB, zC);
}

// out = act( D * Ah^T * D * (Xin @ W) + b ),  Ah = Aadj + 2I (improved GCN)
static void tgun_gcn(const TgunCtx& cx, const float* Aadj, int n,
                     const float* Xin, int cin, const float* W, const float* b,
                     int cout, int relu, float* Xout)
{
    // T0 = Xin @ W
    tgun_gemm(cx.stream, Xin, W, cx.T0, n, cout, cin, cin, cout, cout, 0, 0, 0, 0, 0);
    tgun_coldeg_dis<<<gsz(n), 256, 0, cx.stream>>>(Aadj, cx.DIS, n);
    // T1 = dis ⊙ T0
    tgun_rowscale<<<gsz((size_t)n * cout), 256, 0, cx.stream>>>(cx.T0, cx.DIS, cx.T1, n, cout);
    // T0 = A^T @ T1
    tgun_gemm(cx.stream, Aadj, cx.T1, cx.T0, n, cout, n, n, cout, cout, 1, 0, 0, 0, 0);
    // Xout = act(dis ⊙ (T0 + 2*T1) + b)
    tgun_gcn_epi<<<gsz((size_t)n * cout), 256, 0, cx.stream>>>(cx.T0, cx.T1, cx.DIS, b,
                                                               Xout, n, cout, relu);
}

extern "C" void kernel_launch(void* const* d_in, const int* in_sizes, int n_in,
                              void* d_out, int out_size, void* d_ws, size_t ws_size,
                              hipStream_t stream)
{
    const float* x     = (const float*)d_in[0];
    const int*   eidx  = (const int*)  d_in[1];
    const float* W0    = (const float*)d_in[2];
    const float* b0    = (const float*)d_in[3];
    const float* Wd    = (const float*)d_in[4];
    const float* bd    = (const float*)d_in[5];
    const float* pvec  = (const float*)d_in[6];
    const float* Wih   = (const float*)d_in[7];
    const float* Whh   = (const float*)d_in[8];
    const float* bih   = (const float*)d_in[9];
    const float* bhh   = (const float*)d_in[10];
    const float* Wu    = (const float*)d_in[11];
    const float* bu    = (const float*)d_in[12];
    const float* Wlast = (const float*)d_in[13];
    const float* blast = (const float*)d_in[14];

    const int E = in_sizes[1] / 2;
    const int H = H_DIM;

    // -------- workspace layout (floats), ~189 MB total --------
    float* w = (float*)d_ws;
    size_t off = 0;
    float* A0   = w + off; off += (size_t)4096 * 4096;   // As[0], preserved
    float* AAUG = w + off; off += (size_t)4096 * 4096;   // augment scratch
    float* A1   = w + off; off += (size_t)2048 * 2048;   // As[1]
    float* A2   = w + off; off += (size_t)1024 * 1024;   // As[2]
    float* A3   = w + off; off += (size_t)512  * 512;    // level-2 pooled A
    float* X0   = w + off; off += (size_t)4096 * H;      // xs[0]
    float* X1   = w + off; off += (size_t)2048 * H;      // xs[1]
    float* X2   = w + off; off += (size_t)1024 * H;      // xs[2]
    float* XA   = w + off; off += (size_t)4096 * H;      // ping
    float* XB   = w + off; off += (size_t)4096 * H;      // pong
    float* T0   = w + off; off += (size_t)4096 * H;
    float* T1   = w + off; off += (size_t)4096 * H;
    float* G    = w + off; off += (size_t)2048 * 4 * H;  // LSTM gate precompute
    float* SC   = w + off; off += 4096;                  // scores (tanh'ed)
    float* DIS  = w + off; off += 4096;
    float* PN   = w + off; off += 16;                    // ||p||
    int* PERM0  = (int*)(w + off); off += 2048;
    int* PERM1  = (int*)(w + off); off += 1024;
    int* PERM2  = (int*)(w + off); off += 512;
    (void)ws_size; (void)n_in; (void)out_size;

    TgunCtx cx{stream, T0, T1, DIS};

    const int   nlev[4] = {4096, 2048, 1024, 512};
    float*      Al[4]   = {A0, A1, A2, A3};
    float*      Xs[3]   = {X0, X1, X2};
    int*        Pm[3]   = {PERM0, PERM1, PERM2};

    // -------- build dense adjacency (coalesced duplicate edges) --------
    hipMemsetAsync(A0, 0, (size_t)4096 * 4096 * sizeof(float), stream);
    tgun_scatter_edges<<<gsz(E), 256, 0, stream>>>(eidx, A0, E, 4096);

    // -------- stem GCN: xs[0] = relu(gcn(A, x, W0, b0)) --------
    tgun_gcn(cx, A0, 4096, x, C_IN, W0, b0, H, 1, X0);

    // -------- encoder: augment -> topk pool -> GCN -> LSTM --------
    float* xcur = X0;
    for (int i = 0; i < DEPTH_C; ++i) {
        int n = nlev[i], k = nlev[i + 1];

        // AAUG = (A|diag=1) @ (A|diag=1), diag zeroed  -- dominant WMMA matmul
        tgun_gemm(stream, Al[i], Al[i], AAUG, n, n, n, n, n, n, 0, 0, 1, 1, 1);

        // top-k pooling (stable descending rank == jax.lax.top_k)
        tgun_pnorm<<<1, 256, 0, stream>>>(pvec + (size_t)i * H, PN, H);
        tgun_score<<<gsz(n), 256, 0, stream>>>(xcur, pvec + (size_t)i * H, PN, SC, n, H);
        tgun_rank_topk<<<gsz(n), 256, 0, stream>>>(SC, Pm[i], n, k);

        float* P = (xcur == XA) ? XB : XA;        // pooled features
        tgun_gather_x<<<gsz((size_t)k * H), 256, 0, stream>>>(xcur, SC, Pm[i], P, k, H);
        tgun_gather_A<<<gsz((size_t)k * k), 256, 0, stream>>>(AAUG, Pm[i], Al[i + 1], k, n);

        float* Q = (P == XA) ? XB : XA;
        tgun_gcn(cx, Al[i + 1], k, P, H, Wd + (size_t)i * H * H, bd + (size_t)i * H,
                 H, 1, Q);

        // LSTM: Gx = Q @ Wih^T (WMMA), then sequential scan -> P
        tgun_gemm(stream, Q, Wih + (size_t)i * 4 * H * H, G,
                  k, 4 * H, H, H, H, 4 * H, 0, 1, 0, 0, 0);
        tgun_lstm_scan<<<1, 1024, 0, stream>>>(G, Whh + (size_t)i * 4 * H * H,
                                               bih + (size_t)i * 4 * H,
                                               bhh + (size_t)i * 4 * H, P, k);
        xcur = P;
        if (i < DEPTH_C - 1)
            tgun_copy<<<gsz((size_t)k * H), 256, 0, stream>>>(xcur, Xs[i + 1], (size_t)k * H);
    }

    // -------- decoder: unpool(sum residual) -> GCN --------
    for (int i = 0; i < DEPTH_C; ++i) {
        int j  = DEPTH_C - 1 - i;
        int nj = nlev[j], kj = nlev[j + 1];
        float* xnew = (xcur == XA) ? XB : XA;
        tgun_copy<<<gsz((size_t)nj * H), 256, 0, stream>>>(Xs[j], xnew, (size_t)nj * H);
        tgun_unpool_add<<<gsz((size_t)kj * H), 256, 0, stream>>>(Pm[j], xcur, xnew, kj, H);
        if (i < DEPTH_C - 1) {
            float* ob = (xnew == XA) ? XB : XA;
            tgun_gcn(cx, Al[j], nj, xnew, H, Wu + (size_t)i * H * H, bu + (size_t)i * H,
                     H, 1, ob);
            xcur = ob;
        } else {
            tgun_gcn(cx, Al[0], nlev[0], xnew, H, Wlast, blast, C_OUT, 0, (float*)d_out);
        }
    }
}